// RegionLoss_14242111553551
// MI455X (gfx1250) — compile-verified
//
#include <hip/hip_runtime.h>
#include <math.h>

#define NBATCH 16
#define NANC   5
#define NKP    9
#define NCLS   13
#define NH_    38
#define NW_    38
#define MAXT_  50
#define NLAB   21                      // 2*NKP+3
#define NPIX   (NH_*NW_)               // 1444
#define NANCH  (NANC*NPIX)             // 7220
#define NTOT   (NBATCH*NANCH)          // 115520
#define TPAD   64
#define NCH    (NANC*(2*NKP+1+NCLS))   // 160
#define CTILES ((NANCH+15)/16)         // 452
#define TTILES (TPAD/16)               // 4

#define TH_V     80.0f
#define SHARP_V  2.0f
#define IMW_V    640.0f
#define IMH_V    480.0f
#define SIL_V    0.6f
#define OBJ_S    5.0f
#define NOOBJ_S  1.0f
#define PRETRAIN_E 15
#define INV9     (1.0f/9.0f)
#define INVDEN   (1.0f/6.3890560989306495f)   // 1/(e^2 - 1)

typedef float v2f __attribute__((ext_vector_type(2)));
typedef float v8f __attribute__((ext_vector_type(8)));

__device__ __forceinline__ float sig_(float v){ return 1.0f/(1.0f+__expf(-v)); }
__device__ __forceinline__ float cconf_(float dist){
  return (dist < TH_V) ? (__expf(SHARP_V*(1.0f - dist*(1.0f/TH_V))) - 1.0f)*INVDEN : 0.0f;
}

// ---------------------------------------------------------------------------
// K0a: valid[b][t] = cumprod_t( gt_x0 != 0 ), padded to TPAD with zeros.
__global__ void k_valid(const float* __restrict__ tgt, float* __restrict__ valid){
  int b = threadIdx.x;
  if (b >= NBATCH) return;
  float v = 1.0f;
  for (int t = 0; t < TPAD; ++t){
    if (t < MAXT_){
      float gx0 = tgt[(b*MAXT_ + t)*NLAB + 1];
      v *= (gx0 != 0.0f) ? 1.0f : 0.0f;
    } else v = 0.0f;
    valid[b*TPAD + t] = v;
  }
}

// ---------------------------------------------------------------------------
// K0b: B-matrix table {tx, ty, tx^2+ty^2}, cell->target init (-1), zero loss.
__global__ void k_prep(const float* __restrict__ tgt, float* __restrict__ btab,
                       int* __restrict__ cellt, float* __restrict__ loss){
  int idx = blockIdx.x*blockDim.x + threadIdx.x;
  if (idx == 0) loss[0] = 0.0f;
  if (idx < NTOT) cellt[idx] = -1;
  if (idx < NBATCH*TPAD*NKP){
    int b = idx / (TPAD*NKP);
    int r = idx - b*(TPAD*NKP);
    int t = r / NKP;
    int k = r - t*NKP;
    float tx = 0.0f, ty = 0.0f;
    if (t < MAXT_){
      tx = tgt[(b*MAXT_ + t)*NLAB + 1 + 2*k] * IMW_V;
      ty = tgt[(b*MAXT_ + t)*NLAB + 2 + 2*k] * IMH_V;
    }
    btab[idx*4 + 0] = tx;
    btab[idx*4 + 1] = ty;
    btab[idx*4 + 2] = tx*tx + ty*ty;
    btab[idx*4 + 3] = 0.0f;
  }
}

// ---------------------------------------------------------------------------
// K1: WMMA pairwise-distance confidence. One wave (32 lanes) per block.
// dist^2[M,N] = [sx^2+sy^2, -2sx, -2sy, 1] . [1, tx, ty, tx^2+ty^2]
// via V_WMMA_F32_16X16X4_F32; transcendental tail on VALU; cross-lane max.
__global__ void __launch_bounds__(32) k_conf(const float* __restrict__ outp,
                                             const float* __restrict__ btab,
                                             const float* __restrict__ valid,
                                             float* __restrict__ noobj){
  const int b    = blockIdx.x / CTILES;
  const int tile = blockIdx.x % CTILES;
  const int lane = threadIdx.x;          // 0..31
  const int m    = lane & 15;            // cell-in-tile / target-in-tile
  const bool hi  = lane >= 16;

  __builtin_prefetch(btab + (size_t)(b*TPAD)*NKP*4, 0, 1);

  int cellr = tile*16 + m;
  float pad = (cellr < NANCH) ? 0.0f : 1.0e7f;  // force dist>TH for pad cells
  int cell  = (cellr < NANCH) ? cellr : (NANCH-1);
  int a   = cell / NPIX;
  int rem = cell - a*NPIX;
  int gj  = rem / NW_;
  int gi  = rem - gj*NW_;
  const float* ob = outp + (size_t)(b*NCH + a*32)*NPIX + rem;

  // A-matrix rows per corner: lanes 0-15 hold K=0,1 ; lanes 16-31 hold K=2,3
  v2f Amat[NKP];
  #pragma unroll
  for (int k = 0; k < NKP; ++k){
    float xv = ob[(2*k  )*NPIX];
    float yv = ob[(2*k+1)*NPIX];
    if (k == 0){ xv = sig_(xv); yv = sig_(yv); }
    float sx = (xv + (float)gi) * (IMW_V/(float)NW_) + pad;
    float sy = (yv + (float)gj) * (IMH_V/(float)NH_) + pad;
    float a0 = hi ? (-2.0f*sy) : (sx*sx + sy*sy);
    float a1 = hi ? 1.0f       : (-2.0f*sx);
    v2f av; av.x = a0; av.y = a1;
    Amat[k] = av;
  }

  float curmax[8];
  #pragma unroll
  for (int v = 0; v < 8; ++v) curmax[v] = 0.0f;

  for (int tt = 0; tt < TTILES; ++tt){
    int t = tt*16 + m;
    const float* bb = btab + (size_t)((b*TPAD + t)*NKP)*4;
    float vt = valid[b*TPAD + t];
    float csum[8];
    #pragma unroll
    for (int v = 0; v < 8; ++v) csum[v] = 0.0f;

    #pragma unroll
    for (int k = 0; k < NKP; ++k){
      float tx  = bb[k*4 + 0];
      float ty  = bb[k*4 + 1];
      float tn2 = bb[k*4 + 2];
      // B cols: lanes 0-15 hold K=0,1 ; lanes 16-31 hold K=2,3
      v2f Bmat; Bmat.x = hi ? ty : 1.0f; Bmat.y = hi ? tn2 : tx;
      v8f cz = {};
      v8f d = __builtin_amdgcn_wmma_f32_16x16x4_f32(
                  false, Amat[k], false, Bmat, (short)0, cz, false, false);
      #pragma unroll
      for (int v = 0; v < 8; ++v){
        float dd = d[v];
        dd = dd > 0.0f ? dd : 0.0f;
        csum[v] += cconf_(sqrtf(dd));
      }
    }
    #pragma unroll
    for (int v = 0; v < 8; ++v)
      curmax[v] = fmaxf(curmax[v], csum[v]*INV9*vt);
  }

  // max over targets: reduce across the 16 lanes of each half (same M rows)
  #pragma unroll
  for (int off = 1; off < 16; off <<= 1){
    #pragma unroll
    for (int v = 0; v < 8; ++v)
      curmax[v] = fmaxf(curmax[v], __shfl_xor(curmax[v], off, 32));
  }
  if (lane == 0 || lane == 16){
    int mbase = tile*16 + (hi ? 8 : 0);
    #pragma unroll
    for (int v = 0; v < 8; ++v){
      int c2 = mbase + v;
      if (c2 < NANCH)
        noobj[b*NANCH + c2] = (curmax[v] > SIL_V) ? 0.0f : NOOBJ_S;
    }
  }
}

// ---------------------------------------------------------------------------
// K2: per-target anchor argmax, tconf (with reference's wrapped pidx), scatter.
__global__ void k_targets(const float* __restrict__ outp, const float* __restrict__ tgt,
                          const float* __restrict__ anchors, const float* __restrict__ valid,
                          float* __restrict__ tconf, int* __restrict__ tmeta,
                          int* __restrict__ cellt){
  int idx = blockIdx.x*blockDim.x + threadIdx.x;
  if (idx >= NBATCH*MAXT_) return;
  int b = idx / MAXT_;
  int t = idx - b*MAXT_;
  const float* tr = tgt + (size_t)(b*MAXT_ + t)*NLAB;

  int gi0 = (int)floorf(tr[1] * (float)NW_);
  int gj0 = (int)floorf(tr[2] * (float)NH_);
  float gw = tr[NLAB-2] * (float)NW_;
  float gh = tr[NLAB-1] * (float)NH_;

  int best = 0; float biou = -1.0f;
  for (int n = 0; n < NANC; ++n){
    float aw = anchors[2*n], ah = anchors[2*n+1];
    float cw = fminf(gw, aw), ch = fminf(gh, ah);
    float ca = cw*ch;
    float ua = gw*gh + aw*ah - ca;
    float iou = (cw > 0.0f && ch > 0.0f) ? ca/ua : 0.0f;
    if (iou > biou){ biou = iou; best = n; }
  }

  // pidx = (b*nAnch - nPix + gj0*nW + gi0) mod (nB*nAnch)  (python mod)
  long long P = (long long)b*NANCH - NPIX + (long long)gj0*NW_ + gi0;
  long long M = (long long)NBATCH*NANCH;
  long long p = ((P % M) + M) % M;
  int b2 = (int)(p / NANCH);
  int r  = (int)(p - (long long)b2*NANCH);
  int a2 = r / NPIX;
  int c2 = r - a2*NPIX;
  int j2 = c2 / NW_;
  int i2 = c2 - j2*NW_;
  const float* ob = outp + (size_t)(b2*NCH + a2*32)*NPIX + c2;

  float s = 0.0f;
  #pragma unroll
  for (int k = 0; k < NKP; ++k){
    float xv = ob[(2*k  )*NPIX];
    float yv = ob[(2*k+1)*NPIX];
    if (k == 0){ xv = sig_(xv); yv = sig_(yv); }
    float px = (xv + (float)i2) * (1.0f/(float)NW_);
    float py = (yv + (float)j2) * (1.0f/(float)NH_);
    float dx = (tr[1 + 2*k] - px) * IMW_V;
    float dy = (tr[2 + 2*k] - py) * IMH_V;
    s += cconf_(sqrtf(dx*dx + dy*dy));
  }
  tconf[idx] = s * INV9;

  int lab = (int)tr[0];
  lab = lab < 0 ? 0 : (lab > NCLS-1 ? NCLS-1 : lab);
  tmeta[idx*4 + 0] = gi0;
  tmeta[idx*4 + 1] = gj0;
  tmeta[idx*4 + 2] = lab;
  tmeta[idx*4 + 3] = best;

  if (valid[b*TPAD + t] > 0.0f){
    int flat = ((b*NANC + best)*NH_ + gj0)*NW_ + gi0;
    flat = flat < 0 ? 0 : (flat >= NTOT ? NTOT-1 : flat);
    atomicMax(&cellt[flat], t);   // deterministic "last t wins"
  }
}

// ---------------------------------------------------------------------------
// K3: per-cell loss accumulation + hierarchical reduction.
__global__ void k_loss(const float* __restrict__ outp, const float* __restrict__ tgt,
                       const float* __restrict__ noobj, const float* __restrict__ tconf,
                       const int* __restrict__ tmeta, const int* __restrict__ cellt,
                       const int* __restrict__ epoch, float* __restrict__ loss){
  int idx = blockIdx.x*blockDim.x + threadIdx.x;
  float part = 0.0f;
  if (idx < NTOT){
    int i = idx % NW_;
    int j = (idx / NW_) % NH_;
    int a = (idx / NPIX) % NANC;
    int b = idx / (NANC*NPIX);
    const float* ob = outp + (size_t)(b*NCH + a*32)*NPIX + j*NW_ + i;
    float conf = sig_(ob[(2*NKP)*NPIX]);

    int tw = cellt[idx];
    float scale, tc;
    if (tw >= 0){
      int ti = b*MAXT_ + tw;
      scale = OBJ_S;
      tc = tconf[ti];
      const float* tr = tgt + (size_t)ti*NLAB;
      int gi0 = tmeta[ti*4 + 0];
      int gj0 = tmeta[ti*4 + 1];
      int lab = tmeta[ti*4 + 2];
      float lxy = 0.0f;
      #pragma unroll
      for (int k = 0; k < NKP; ++k){
        float xv = ob[(2*k  )*NPIX];
        float yv = ob[(2*k+1)*NPIX];
        if (k == 0){ xv = sig_(xv); yv = sig_(yv); }
        float txk = tr[1 + 2*k]*(float)NW_ - (float)gi0;
        float tyk = tr[2 + 2*k]*(float)NH_ - (float)gj0;
        lxy += (xv - txk)*(xv - txk) + (yv - tyk)*(yv - tyk);
      }
      part += 0.5f * lxy;                       // COORD_SCALE = 1
      float l[NCLS]; float mx = -1.0e30f;
      #pragma unroll
      for (int c = 0; c < NCLS; ++c){
        l[c] = ob[(2*NKP + 1 + c)*NPIX];
        mx = fmaxf(mx, l[c]);
      }
      float se = 0.0f;
      #pragma unroll
      for (int c = 0; c < NCLS; ++c) se += __expf(l[c] - mx);
      part += (mx + __logf(se)) - l[lab];       // CLASS_SCALE = 1
    } else {
      scale = noobj[b*NANCH + a*NPIX + j*NW_ + i];
      tc = 0.0f;
    }
    if (*epoch > PRETRAIN_E)
      part += 0.5f * scale * (conf - tc)*(conf - tc);
  }

  // wave32 + block reduction
  #pragma unroll
  for (int off = 16; off > 0; off >>= 1)
    part += __shfl_xor(part, off, 32);
  __shared__ float sred[8];
  int w = threadIdx.x >> 5;
  if ((threadIdx.x & 31) == 0) sred[w] = part;
  __syncthreads();
  if (threadIdx.x == 0){
    float s = 0.0f;
    for (int k = 0; k < (int)(blockDim.x >> 5); ++k) s += sred[k];
    atomicAdd(loss, s);
  }
}

// ---------------------------------------------------------------------------
extern "C" void kernel_launch(void* const* d_in, const int* in_sizes, int n_in,
                              void* d_out, int out_size, void* d_ws, size_t ws_size,
                              hipStream_t stream){
  const float* outp    = (const float*)d_in[0];
  const float* target  = (const float*)d_in[1];
  const float* anchors = (const float*)d_in[2];
  const int*   epoch   = (const int*)d_in[3];
  float* loss = (float*)d_out;

  float* w_valid = (float*)d_ws;                       // NBATCH*TPAD
  float* w_btab  = w_valid + NBATCH*TPAD;              // NBATCH*TPAD*NKP*4
  float* w_noobj = w_btab  + NBATCH*TPAD*NKP*4;        // NTOT
  float* w_tconf = w_noobj + NTOT;                     // NBATCH*MAXT_
  int*   w_tmeta = (int*)(w_tconf + NBATCH*MAXT_);     // NBATCH*MAXT_*4
  int*   w_cellt = w_tmeta + NBATCH*MAXT_*4;           // NTOT

  hipLaunchKernelGGL(k_valid, dim3(1), dim3(32), 0, stream, target, w_valid);
  hipLaunchKernelGGL(k_prep, dim3((NTOT+255)/256), dim3(256), 0, stream,
                     target, w_btab, w_cellt, loss);
  hipLaunchKernelGGL(k_conf, dim3(NBATCH*CTILES), dim3(32), 0, stream,
                     outp, w_btab, w_valid, w_noobj);
  hipLaunchKernelGGL(k_targets, dim3((NBATCH*MAXT_+255)/256), dim3(256), 0, stream,
                     outp, target, anchors, w_valid, w_tconf, w_tmeta, w_cellt);
  hipLaunchKernelGGL(k_loss, dim3((NTOT+255)/256), dim3(256), 0, stream,
                     outp, target, w_noobj, w_tconf, w_tmeta, w_cellt, epoch, loss);
}